// Attentions_14809047237038
// MI455X (gfx1250) — compile-verified
//
#include <hip/hip_runtime.h>
#include <math.h>

// CDNA5 / gfx1250 flash-attention for B=4,H=8,S=2048,D=3 (fp32).
// Scores via V_WMMA_F32_16X16X4_F32 computing S^T = K * Q^T so each lane owns
// one query column -> lane-local online softmax + P*V accumulation.
// v2: K stored in LDS pre-swizzled to the WMMA A-operand layout (one aligned
//     ds_load_b64, zero cndmasks), software-pipelined A prefetch, and two
//     query tiles per wave sharing K/V loads (2 independent WMMA chains).

typedef float v2f __attribute__((ext_vector_type(2)));
typedef float v8f __attribute__((ext_vector_type(8)));

#define B_    4
#define H_    8
#define S_    2048
#define D_    3
#define WAVES 8
#define BLK   (WAVES * 32)
#define NT    (S_ / 16)          // 128 key tiles
#define QTPW  2                  // query tiles per wave

__device__ __forceinline__ float fast_exp2(float x) {
#if __has_builtin(__builtin_amdgcn_exp2f)
  return __builtin_amdgcn_exp2f(x);   // v_exp_f32
#else
  return exp2f(x);
#endif
}

__global__ __launch_bounds__(BLK) void attn_f32_wmma_kernel(
    const float* __restrict__ x,    // [B,1,S,3]
    const float* __restrict__ Wq,   // [1,H,3,3]
    const float* __restrict__ Wk,   // [1,H,3,3]
    const float* __restrict__ Wv,   // [1,H,3,3]
    float* __restrict__ out)        // 2x [B,H,S,3] concatenated
{
  // K rows pre-swizzled for the A operand: [k0,k1,k2,0] per key (16B).
  __shared__ __align__(16) float lKA[S_ * 4];        // 32 KB
  __shared__ __align__(16) float lV [S_ * 4];        // 32 KB

  const int bh   = blockIdx.x;
  const int b    = bh / H_;
  const int h    = bh % H_;
  const int tid  = threadIdx.x;
  const int lane = tid & 31;
  const int wave = tid >> 5;
  const bool hiHalf = (lane >= 16);

  const float* xb = x  + (size_t)b * (S_ * D_);
  const float* wq = Wq + h * 9;
  const float* wk = Wk + h * 9;
  const float* wv = Wv + h * 9;

  // ---- Stage K (A-operand layout) and V into LDS (block cooperates) ----
  #pragma unroll
  for (int j = 0; j < S_ / BLK; ++j) {
    const int k = j * BLK + tid;
    const float x0 = xb[k * 3 + 0];
    const float x1 = xb[k * 3 + 1];
    const float x2 = xb[k * 3 + 2];
    lKA[k * 4 + 0] = x0 * wk[0] + x1 * wk[3] + x2 * wk[6];
    lKA[k * 4 + 1] = x0 * wk[1] + x1 * wk[4] + x2 * wk[7];
    lKA[k * 4 + 2] = x0 * wk[2] + x1 * wk[5] + x2 * wk[8];
    lKA[k * 4 + 3] = 0.0f;                       // zero-pad for hi-half a.y
    lV [k * 4 + 0] = x0 * wv[0] + x1 * wv[3] + x2 * wv[6];
    lV [k * 4 + 1] = x0 * wv[1] + x1 * wv[4] + x2 * wv[7];
    lV [k * 4 + 2] = x0 * wv[2] + x1 * wv[5] + x2 * wv[8];
    lV [k * 4 + 3] = 0.0f;
  }
  __syncthreads();

  // ---- Two query tiles per wave: Q projections, pre-scaled ----
  const int pair  = blockIdx.y * WAVES + wave;       // 0..63
  const int qrow0 = pair * 32 + (lane & 15);
  const int qrow1 = qrow0 + 16;
  const float qs  = 1.4426950408889634f * 0.5773502691896258f; // log2e/sqrt(3)

  v2f bOp0, bOp1;
  {
    const float a = xb[qrow0 * 3 + 0], c = xb[qrow0 * 3 + 1], e = xb[qrow0 * 3 + 2];
    const float Q0 = (a * wq[0] + c * wq[3] + e * wq[6]) * qs;
    const float Q1 = (a * wq[1] + c * wq[4] + e * wq[7]) * qs;
    const float Q2 = (a * wq[2] + c * wq[5] + e * wq[8]) * qs;
    bOp0.x = hiHalf ? Q2 : Q0;
    bOp0.y = hiHalf ? 0.0f : Q1;
  }
  {
    const float a = xb[qrow1 * 3 + 0], c = xb[qrow1 * 3 + 1], e = xb[qrow1 * 3 + 2];
    const float Q0 = (a * wq[0] + c * wq[3] + e * wq[6]) * qs;
    const float Q1 = (a * wq[1] + c * wq[4] + e * wq[7]) * qs;
    const float Q2 = (a * wq[2] + c * wq[5] + e * wq[8]) * qs;
    bOp1.x = hiHalf ? Q2 : Q0;
    bOp1.y = hiHalf ? 0.0f : Q1;
  }

  // Flash state, lane-local (this lane covers its key half)
  float m0 = -3.0e38f, l0 = 0.0f, p00 = 0.0f, p01 = 0.0f, p02 = 0.0f;
  float m1 = -3.0e38f, l1 = 0.0f, p10 = 0.0f, p11 = 0.0f, p12 = 0.0f;

  const float4* lV4 = (const float4*)lV;
  const int krow16 = lane & 15;
  const int hal8   = hiHalf ? 8 : 0;
  // A-operand float index for this lane within a tile (8B-aligned v2f):
  const int aIdx = krow16 * 4 + (hiHalf ? 2 : 0);

  v2f aCur = *(const v2f*)&lKA[aIdx];                // tile 0 prefetch

  for (int t = 0; t < NT; ++t) {
    // Prefetch next tile's A operand (wraps harmlessly on the last iter).
    const int tn = (t + 1) & (NT - 1);
    const v2f aNext = *(const v2f*)&lKA[tn * 64 + aIdx];

    v8f c0 = {};
    v8f c1 = {};
    c0 = __builtin_amdgcn_wmma_f32_16x16x4_f32(
        false, aCur, false, bOp0, (short)0, c0, false, false);
    c1 = __builtin_amdgcn_wmma_f32_16x16x4_f32(
        false, aCur, false, bOp1, (short)0, c1, false, false);

    float tm0 = c0[0], tm1 = c1[0];
    #pragma unroll
    for (int i = 1; i < 8; ++i) { tm0 = fmaxf(tm0, c0[i]); tm1 = fmaxf(tm1, c1[i]); }
    const float mn0 = fmaxf(m0, tm0);
    const float mn1 = fmaxf(m1, tm1);
    const float cr0 = fast_exp2(m0 - mn0);
    const float cr1 = fast_exp2(m1 - mn1);
    m0 = mn0; m1 = mn1;
    l0 *= cr0; p00 *= cr0; p01 *= cr0; p02 *= cr0;
    l1 *= cr1; p10 *= cr1; p11 *= cr1; p12 *= cr1;

    const int kb = t * 16;
    #pragma unroll
    for (int i = 0; i < 8; ++i) {
      const float4 v = lV4[kb + hal8 + i];   // shared by both query tiles
      const float e0 = fast_exp2(c0[i] - mn0);
      const float e1 = fast_exp2(c1[i] - mn1);
      l0 += e0;                 l1 += e1;
      p00 = fmaf(e0, v.x, p00); p10 = fmaf(e1, v.x, p10);
      p01 = fmaf(e0, v.y, p01); p11 = fmaf(e1, v.y, p11);
      p02 = fmaf(e0, v.z, p02); p12 = fmaf(e1, v.z, p12);
    }
    aCur = aNext;
  }

  // ---- Combine the two key-halves (lane L <-> L^16 share a query) ----
  const size_t dup = (size_t)B_ * H_ * S_ * 3;       // tuple returns (ctx, ctx)
  const size_t headOff = (size_t)(b * H_ + h) * S_;

  {
    const float m_o = __shfl_xor(m0, 16, 32);
    const float l_o = __shfl_xor(l0, 16, 32);
    const float b0  = __shfl_xor(p00, 16, 32);
    const float b1  = __shfl_xor(p01, 16, 32);
    const float b2  = __shfl_xor(p02, 16, 32);
    const float mt = fmaxf(m0, m_o);
    const float fs = fast_exp2(m0 - mt);
    const float fo = fast_exp2(m_o - mt);
    const float inv = 1.0f / (l0 * fs + l_o * fo);
    const float o0 = (p00 * fs + b0 * fo) * inv;
    const float o1 = (p01 * fs + b1 * fo) * inv;
    const float o2 = (p02 * fs + b2 * fo) * inv;
    if (!hiHalf) {
      const size_t off = (headOff + qrow0) * 3;
      out[off + 0] = o0; out[off + 1] = o1; out[off + 2] = o2;
      out[dup + off + 0] = o0; out[dup + off + 1] = o1; out[dup + off + 2] = o2;
    }
  }
  {
    const float m_o = __shfl_xor(m1, 16, 32);
    const float l_o = __shfl_xor(l1, 16, 32);
    const float b0  = __shfl_xor(p10, 16, 32);
    const float b1  = __shfl_xor(p11, 16, 32);
    const float b2  = __shfl_xor(p12, 16, 32);
    const float mt = fmaxf(m1, m_o);
    const float fs = fast_exp2(m1 - mt);
    const float fo = fast_exp2(m_o - mt);
    const float inv = 1.0f / (l1 * fs + l_o * fo);
    const float o0 = (p10 * fs + b0 * fo) * inv;
    const float o1 = (p11 * fs + b1 * fo) * inv;
    const float o2 = (p12 * fs + b2 * fo) * inv;
    if (!hiHalf) {
      const size_t off = (headOff + qrow1) * 3;
      out[off + 0] = o0; out[off + 1] = o1; out[off + 2] = o2;
      out[dup + off + 0] = o0; out[dup + off + 1] = o1; out[dup + off + 2] = o2;
    }
  }
}

extern "C" void kernel_launch(void* const* d_in, const int* in_sizes, int n_in,
                              void* d_out, int out_size, void* d_ws, size_t ws_size,
                              hipStream_t stream) {
  (void)in_sizes; (void)n_in; (void)out_size; (void)d_ws; (void)ws_size;
  const float* x  = (const float*)d_in[0];
  const float* Wq = (const float*)d_in[1];
  const float* Wk = (const float*)d_in[2];
  const float* Wv = (const float*)d_in[3];
  float* out = (float*)d_out;

  dim3 grid(B_ * H_, (S_ / 16) / (WAVES * QTPW));   // (32, 8)
  attn_f32_wmma_kernel<<<grid, BLK, 0, stream>>>(x, Wq, Wk, Wv, out);
}